// Model_7009386627748
// MI455X (gfx1250) — compile-verified
//
#include <hip/hip_runtime.h>
#include <stddef.h>

typedef float v2f __attribute__((ext_vector_type(2)));
typedef float v4f __attribute__((ext_vector_type(4)));
typedef float v8f __attribute__((ext_vector_type(8)));

#define LENS   65536
#define KDIM   256
#define NDIM   80
#define NTILES 5          // 80 / 16
#define MTILES (LENS / 16)

// ---------------- prefix scan of `predicted` (exclusive) ----------------

__global__ void scan_block_kernel(const int* __restrict__ pred,
                                  int* __restrict__ offs,
                                  int* __restrict__ blockSums) {
    __shared__ int s[256];
    const int t = threadIdx.x;
    const int g = blockIdx.x * 256 + t;
    const int v = pred[g];
    s[t] = v;
    __syncthreads();
    #pragma unroll
    for (int d = 1; d < 256; d <<= 1) {
        int tmp = (t >= d) ? s[t - d] : 0;
        __syncthreads();
        s[t] += tmp;
        __syncthreads();
    }
    offs[g] = s[t] - v;                    // exclusive scan within block
    if (t == 255) blockSums[blockIdx.x] = s[255];
}

__global__ void scan_sums_kernel(int* __restrict__ blockSums) {
    __shared__ int s[256];
    const int t = threadIdx.x;
    const int v = blockSums[t];
    s[t] = v;
    __syncthreads();
    #pragma unroll
    for (int d = 1; d < 256; d <<= 1) {
        int tmp = (t >= d) ? s[t - d] : 0;
        __syncthreads();
        s[t] += tmp;
        __syncthreads();
    }
    blockSums[t] = s[t] - v;               // exclusive
}

__global__ void add_base_kernel(int* __restrict__ offs,
                                const int* __restrict__ blockSums) {
    offs[blockIdx.x * 256 + threadIdx.x] += blockSums[blockIdx.x];
}

// ---------------- WMMA GEMM: x = inp @ W^T + b  ([65536,256]x[80,256]^T) ----------------
// One wave computes a 16x80 strip using V_WMMA_F32_16X16X4_F32.
// A 16x4 layout : lane(lo,hi) VGPR v = inp[m0+lo][k0 + 2*hi + v]
// B 4x16 layout : lane(lo,hi) VGPR v = W[n0+lo][k0 + 2*hi + v]   (B[k][n] = W[n][k])
// C/D layout    : lane(lo,hi) VGPR v = x[m0 + v + 8*hi][n0 + lo]

__global__ void gemm_wmma_kernel(const float* __restrict__ inp,
                                 const float* __restrict__ W,
                                 const float* __restrict__ bias,
                                 float* __restrict__ x) {
    const int lane  = threadIdx.x & 31;
    const int wave  = threadIdx.x >> 5;
    const int mtile = blockIdx.x * 8 + wave;     // 8 waves / block
    const int m0    = mtile * 16;
    const int lo    = lane & 15;
    const int hi    = lane >> 4;

    v8f acc[NTILES];
    #pragma unroll
    for (int t = 0; t < NTILES; ++t) acc[t] = (v8f){0.f,0.f,0.f,0.f,0.f,0.f,0.f,0.f};

    const float* arow = inp + (size_t)(m0 + lo) * KDIM + 2 * hi;

    for (int k0 = 0; k0 < KDIM; k0 += 4) {
        const v2f a = *(const v2f*)(arow + k0);
        #pragma unroll
        for (int t = 0; t < NTILES; ++t) {
            const v2f bv = *(const v2f*)(W + (size_t)(16 * t + lo) * KDIM + k0 + 2 * hi);
            acc[t] = __builtin_amdgcn_wmma_f32_16x16x4_f32(
                /*neg_a=*/false, a, /*neg_b=*/false, bv,
                /*c_mod=*/(short)0, acc[t], /*reuse_a=*/false, /*reuse_b=*/false);
        }
    }

    #pragma unroll
    for (int t = 0; t < NTILES; ++t) {
        const float bb = bias[16 * t + lo];
        #pragma unroll
        for (int v = 0; v < 8; ++v) {
            const int m = m0 + v + 8 * hi;
            x[(size_t)m * NDIM + 16 * t + lo] = acc[t][v] + bb;
        }
    }
}

// ---------------- variable row-repeat expansion ----------------
// One wave per source row. Lanes 0..19 hold one float4 of the 80-float row,
// replayed predicted[i] times with nontemporal (streaming) stores.

__global__ void expand_kernel(const float* __restrict__ x,
                              const int* __restrict__ pred,
                              const int* __restrict__ offs,
                              float* __restrict__ out) {
    const int row  = (blockIdx.x * blockDim.x + threadIdx.x) >> 5;
    const int lane = threadIdx.x & 31;
    if (row >= LENS) return;

    const int r    = pred[row];
    const int base = offs[row];

    if (lane < 20) {
        const v4f v = ((const v4f*)(x + (size_t)row * NDIM))[lane];
        for (int j = 0; j < r; ++j) {
            v4f* dst = ((v4f*)(out + (size_t)(base + j) * NDIM)) + lane;
            __builtin_nontemporal_store(v, dst);
        }
    }
}

// ---------------- host launcher ----------------

extern "C" void kernel_launch(void* const* d_in, const int* in_sizes, int n_in,
                              void* d_out, int out_size, void* d_ws, size_t ws_size,
                              hipStream_t stream) {
    const float* inp  = (const float*)d_in[0];   // [1, 65536, 256]
    const float* W    = (const float*)d_in[1];   // [80, 256]
    const float* bias = (const float*)d_in[2];   // [80]
    const int*   pred = (const int*)d_in[3];     // [65536]
    float*       out  = (float*)d_out;           // [total_len, 80]

    char* ws = (char*)d_ws;
    float* x         = (float*)ws;                                   // 65536*80 f32
    int*   offs      = (int*)(ws + (size_t)LENS * NDIM * sizeof(float));
    int*   blockSums = (int*)(ws + (size_t)LENS * NDIM * sizeof(float)
                                 + (size_t)LENS * sizeof(int));

    // exclusive prefix sum of predicted -> offs
    scan_block_kernel<<<LENS / 256, 256, 0, stream>>>(pred, offs, blockSums);
    scan_sums_kernel<<<1, 256, 0, stream>>>(blockSums);
    add_base_kernel<<<LENS / 256, 256, 0, stream>>>(offs, blockSums);

    // x = inp @ W^T + b   (WMMA, one wave per 16x80 strip, 8 waves/block)
    gemm_wmma_kernel<<<MTILES / 8, 256, 0, stream>>>(inp, W, bias, x);

    // repeat-expand rows into out (one wave per source row, 8 rows/block)
    expand_kernel<<<LENS / 8, 256, 0, stream>>>(x, pred, offs, out);
}